// EmbeddingPropagationLayer_1984274891309
// MI455X (gfx1250) — compile-verified
//
#include <hip/hip_runtime.h>
#include <math.h>

typedef __attribute__((ext_vector_type(2))) float v2f;
typedef __attribute__((ext_vector_type(8))) float v8f;

#define NN 8192
#define DD 64
#define BM 128           // output rows per block (8 waves x 16)
#define KC 32            // K chunk staged in LDS
#define SA (KC + 2)      // padded LDS stride (even -> 8B-aligned b64 loads)
#define ALPHA 0.2f
#define EPS 1e-8f

// ---------------- normalization prep ----------------

__global__ __launch_bounds__(256) void rowsum_kernel(const float* __restrict__ adj,
                                                     float* __restrict__ sr) {
  __shared__ float red[256];
  const int row = blockIdx.x;
  const int t = threadIdx.x;
  const float* p = adj + (size_t)row * NN;
  float s = 0.f;
  for (int j = t; j < NN; j += 256) s += p[j];
  red[t] = s;
  __syncthreads();
  for (int w = 128; w > 0; w >>= 1) {
    if (t < w) red[t] += red[t + w];
    __syncthreads();
  }
  if (t == 0) sr[row] = sqrtf(red[0]);
}

__global__ __launch_bounds__(256) void zero_kernel(float* __restrict__ p, int n) {
  int i = blockIdx.x * 256 + threadIdx.x;
  if (i < n) p[i] = 0.f;
}

__global__ __launch_bounds__(256) void colsum_partial_kernel(const float* __restrict__ adj,
                                                             float* __restrict__ acc) {
  const int j = blockIdx.x * 256 + threadIdx.x;
  const int i0 = blockIdx.y * 128;
  float s = 0.f;
#pragma unroll 4
  for (int i = 0; i < 128; ++i) s += adj[(size_t)(i0 + i) * NN + j];
  atomicAdd(&acc[j], s);
}

__global__ __launch_bounds__(256) void sqrt_kernel(const float* __restrict__ a,
                                                   float* __restrict__ sc) {
  int j = blockIdx.x * 256 + threadIdx.x;
  sc[j] = sqrtf(a[j]);
}

// ---------------- big WMMA GEMM: Z1 = Mn @ Bsrc, Z2 = Mn @ Btgt ----------------
// Mn = normalized adjacency (TRANSA=false) or its transpose (TRANSA=true).
// Normalization Mn[i,u] = adj[i,u] / (sr[i]*sc[u] + EPS) fused into the tile load.

template <bool TRANSA>
__global__ __launch_bounds__(256) void spmm_kernel(
    const float* __restrict__ adj, const float* __restrict__ sr, const float* __restrict__ sc,
    const float* __restrict__ Bsrc, const float* __restrict__ Btgt,
    float* __restrict__ Z1, float* __restrict__ Z2) {
  __shared__ __align__(16) float At[BM * SA];   // [row][k]
  __shared__ __align__(16) float Bs[DD * SA];   // [col][k] (transposed src tile)
  __shared__ __align__(16) float Bt[DD * SA];   // [col][k] (transposed tgt tile)

  const int tid = threadIdx.x;
  const int wave = tid >> 5;
  const int lane = tid & 31;
  const int laneRow = lane & 15;
  const int laneHi = lane >> 4;
  const int r0 = blockIdx.x * BM;

  v8f acc[8] = {};   // 4 col-tiles of Z1, then 4 col-tiles of Z2

  for (int k0 = 0; k0 < NN; k0 += KC) {
    if (!TRANSA) {
      // At[row][k] = adjn[r0+row, k0+k]; coalesced float4 along k
#pragma unroll
      for (int it = 0; it < (BM * KC / 4) / 256; ++it) {
        int f = it * 256 + tid;
        int row = f >> 3;            // KC/4 = 8 float4 per row
        int k = (f & 7) << 2;
        float4 v = *(const float4*)(adj + (size_t)(r0 + row) * NN + (k0 + k));
        float a = sr[r0 + row];
        float* dst = &At[row * SA + k];
        dst[0] = v.x * __builtin_amdgcn_rcpf(a * sc[k0 + k + 0] + EPS);
        dst[1] = v.y * __builtin_amdgcn_rcpf(a * sc[k0 + k + 1] + EPS);
        dst[2] = v.z * __builtin_amdgcn_rcpf(a * sc[k0 + k + 2] + EPS);
        dst[3] = v.w * __builtin_amdgcn_rcpf(a * sc[k0 + k + 3] + EPS);
      }
    } else {
      // At[row][k] = adjn[k0+k, r0+row]; coalesced float4 along adj's row, transposed store
#pragma unroll
      for (int it = 0; it < (BM * KC / 4) / 256; ++it) {
        int f = it * 256 + tid;
        int k = f >> 5;              // BM/4 = 32 float4 per adj row
        int rw = (f & 31) << 2;
        float4 v = *(const float4*)(adj + (size_t)(k0 + k) * NN + (r0 + rw));
        float a = sr[k0 + k];
        At[(rw + 0) * SA + k] = v.x * __builtin_amdgcn_rcpf(a * sc[r0 + rw + 0] + EPS);
        At[(rw + 1) * SA + k] = v.y * __builtin_amdgcn_rcpf(a * sc[r0 + rw + 1] + EPS);
        At[(rw + 2) * SA + k] = v.z * __builtin_amdgcn_rcpf(a * sc[r0 + rw + 2] + EPS);
        At[(rw + 3) * SA + k] = v.w * __builtin_amdgcn_rcpf(a * sc[r0 + rw + 3] + EPS);
      }
    }
    // B tiles, stored transposed so WMMA B fragments are contiguous K pairs
#pragma unroll
    for (int it = 0; it < (DD * KC / 4) / 256; ++it) {
      int f = it * 256 + tid;
      int u = f >> 4;                // 16 float4 per embedding row
      int col = (f & 15) << 2;
      float4 v1 = *(const float4*)(Bsrc + (size_t)(k0 + u) * DD + col);
      float4 v2 = *(const float4*)(Btgt + (size_t)(k0 + u) * DD + col);
      Bs[(col + 0) * SA + u] = v1.x; Bs[(col + 1) * SA + u] = v1.y;
      Bs[(col + 2) * SA + u] = v1.z; Bs[(col + 3) * SA + u] = v1.w;
      Bt[(col + 0) * SA + u] = v2.x; Bt[(col + 1) * SA + u] = v2.y;
      Bt[(col + 2) * SA + u] = v2.z; Bt[(col + 3) * SA + u] = v2.w;
    }
    __syncthreads();

    // A 16x4 fragment: lanes 0-15 row M, K={kk,kk+1}; lanes 16-31 K={kk+2,kk+3}
    const float* Arow = &At[(wave * 16 + laneRow) * SA + 2 * laneHi];
#pragma unroll
    for (int kk = 0; kk < KC; kk += 4) {
      v2f a = *(const v2f*)(Arow + kk);
#pragma unroll
      for (int n = 0; n < 8; ++n) {
        const float* Bb = (n < 4) ? Bs : Bt;
        int col = (n & 3) * 16 + laneRow;
        v2f b = *(const v2f*)(Bb + col * SA + kk + 2 * laneHi);
        acc[n] = __builtin_amdgcn_wmma_f32_16x16x4_f32(
            false, a, false, b, (short)0, acc[n], false, false);
      }
    }
    __syncthreads();
  }

  // D layout: VGPR v -> (M = v + 8*laneHi, N = laneRow)
#pragma unroll
  for (int n = 0; n < 8; ++n) {
    float* Z = (n < 4) ? Z1 : Z2;
    const int col = (n & 3) * 16 + laneRow;
#pragma unroll
    for (int v = 0; v < 8; ++v) {
      int row = r0 + wave * 16 + v + 8 * laneHi;
      Z[(size_t)row * DD + col] = acc[n][v];
    }
  }
}

// ---------------- epilogue: out = lrelu(Z1@W1 + (src .* Z2)@W2 + tgt) ----------------

__global__ __launch_bounds__(256) void epilogue_kernel(
    const float* __restrict__ Z1, const float* __restrict__ Z2,
    const float* __restrict__ src, const float* __restrict__ tgt,
    const float* __restrict__ W1, const float* __restrict__ W2,
    float* __restrict__ out) {
  __shared__ float z1s[4][DD];
  __shared__ float ps[4][DD];
  const int tid = threadIdx.x;
  const int r = tid >> 6;          // local row 0..3
  const int c = tid & 63;
  const int row = blockIdx.x * 4 + r;
  const size_t off = (size_t)row * DD + c;
  z1s[r][c] = Z1[off];
  ps[r][c] = src[off] * Z2[off];
  __syncthreads();
  float acc = tgt[off];
#pragma unroll 8
  for (int k = 0; k < DD; ++k)
    acc += z1s[r][k] * W1[k * DD + c] + ps[r][k] * W2[k * DD + c];
  out[off] = (acc >= 0.f) ? acc : ALPHA * acc;
}

// ---------------- launch ----------------

extern "C" void kernel_launch(void* const* d_in, const int* in_sizes, int n_in,
                              void* d_out, int out_size, void* d_ws, size_t ws_size,
                              hipStream_t stream) {
  const float* user = (const float*)d_in[0];   // [8192,64]
  const float* item = (const float*)d_in[1];   // [8192,64]
  const float* adj  = (const float*)d_in[2];   // [8192,8192]
  const float* W1   = (const float*)d_in[3];   // [64,64]
  const float* W2   = (const float*)d_in[4];   // [64,64]

  float* ws = (float*)d_ws;
  const size_t ND = (size_t)NN * DD;
  float* sr   = ws;                 // sqrt(row sums)  [8192]
  float* sc   = ws + NN;            // sqrt(col sums)  [8192]
  float* cacc = ws + 2 * NN;        // col-sum scratch [8192]
  float* Z1u  = ws + 3 * NN;        // A^T @ item
  float* Z2u  = Z1u + ND;           // A^T @ user
  float* Z1i  = Z2u + ND;           // A   @ user
  float* Z2i  = Z1i + ND;           // A   @ item

  float* out_user = (float*)d_out;
  float* out_item = out_user + ND;

  rowsum_kernel<<<NN, 256, 0, stream>>>(adj, sr);
  zero_kernel<<<NN / 256, 256, 0, stream>>>(cacc, NN);
  colsum_partial_kernel<<<dim3(NN / 256, 64), 256, 0, stream>>>(adj, cacc);
  sqrt_kernel<<<NN / 256, 256, 0, stream>>>(cacc, sc);

  // user update: M = A^T  (Z1 = A^T@item, Z2 = A^T@user)
  spmm_kernel<true><<<NN / BM, 256, 0, stream>>>(adj, sr, sc, item, user, Z1u, Z2u);
  // item update: M = A    (Z1 = A@user,   Z2 = A@item)
  spmm_kernel<false><<<NN / BM, 256, 0, stream>>>(adj, sr, sc, user, item, Z1i, Z2i);

  // out_user = lrelu(Z1u@W1 + (item .* Z2u)@W2 + user)
  epilogue_kernel<<<NN / 4, 256, 0, stream>>>(Z1u, Z2u, item, user, W1, W2, out_user);
  // out_item = lrelu(Z1i@W1 + (user .* Z2i)@W2 + item)
  epilogue_kernel<<<NN / 4, 256, 0, stream>>>(Z1i, Z2i, user, item, W1, W2, out_item);
}